// CondensationDiagnostics_13984413516424
// MI455X (gfx1250) — compile-verified
//
#include <hip/hip_runtime.h>

// Problem constants (from reference): N=4096 agents, M=256 clusters, K=32.
#define NN 4096
#define MM 256
#define KK 32
#define W_EPS 1e-6f

// GEMM output S = w^T @ [1 | v | v(x)v] : columns 0 (Z), 1..32 (u), 33..1056 (C)
#define C_TOT 1057
#define SC    1152          // padded row stride = 9 supertiles * 128 cols
#define NSPLIT 4            // split-K factor (deterministic tree reduce after)
#define NCHUNK (NN / NSPLIT)
#define LDS_STR 40          // LDS row stride (floats): 16B aligned, conflict-free

typedef __attribute__((ext_vector_type(2))) float v2f;
typedef __attribute__((ext_vector_type(8))) float v8f;

// ------------------------------------------------------------------
// Kernel 1: v_n = inv(omega_child_n) @ mu_n.  One wave (32 lanes) per n,
// row-per-lane Gaussian elimination in LDS (SPD -> no pivoting needed).
// ------------------------------------------------------------------
__global__ __launch_bounds__(32)
void solve_kernel(const float* __restrict__ oc, const float* __restrict__ mu,
                  float* __restrict__ v) {
  __shared__ float As[32 * 33];
  __shared__ float bs[32];
  __shared__ float xs[32];
  const int n   = blockIdx.x;
  const int tid = threadIdx.x;
  const float* A = oc + (size_t)n * (KK * KK);

  #pragma unroll
  for (int i = 0; i < 32; ++i)
    As[i * 33 + tid] = A[i * 32 + tid];
  bs[tid] = mu[(size_t)n * KK + tid];
  __syncthreads();

  for (int k = 0; k < 32; ++k) {          // forward elimination
    float pb = bs[k];
    if (tid > k) {
      float f = As[tid * 33 + k] / As[k * 33 + k];
      for (int j = k + 1; j < 32; ++j)
        As[tid * 33 + j] -= f * As[k * 33 + j];
      bs[tid] -= f * pb;
    }
    __syncthreads();
  }
  for (int k = 31; k >= 0; --k) {         // back substitution
    if (tid == k) xs[k] = bs[k] / As[k * 33 + k];
    __syncthreads();
    if (tid < k) bs[tid] -= As[tid * 33 + k] * xs[k];
    __syncthreads();
  }
  v[(size_t)n * KK + tid] = xs[tid];
}

// ------------------------------------------------------------------
// Kernel 2: Sp[s] = w[s-slice]^T @ [1 | v | v(x)v] via V_WMMA_F32_16X16X4_F32.
// Block = 256 threads (8 waves); block tile = 32 m x 128 c; wave owns a
// 32m x 16c slab (two accumulators sharing one B fragment).
// B fragments are built branch-free from LDS with sentinel columns
// (Vs[.][32]=1, Vs[.][33]=0), so col 0 -> Z, cols 1..32 -> u, 33..1056 -> C.
// ------------------------------------------------------------------
__global__ __launch_bounds__(256)
void moment_gemm_kernel(const float* __restrict__ W, const float* __restrict__ v,
                        float* __restrict__ Sp) {
  __shared__ float As[32 * LDS_STR];   // masked w^T chunk: [n_local][m_local 0..31]
  __shared__ float Vs[32 * LDS_STR];   // v chunk + sentinels: [n_local][l 0..33]

  const int tid   = threadIdx.x;
  const int lane  = tid & 31;
  const int wave  = tid >> 5;
  const int hi    = lane >> 4;     // lanes 16-31 handle K rows +2 (ISA A/B layout)
  const int cl    = lane & 15;
  const int m0    = blockIdx.x * 32;
  const int ctile = blockIdx.y * 8 + wave;
  const int c     = ctile * 16 + cl;
  const int split = blockIdx.z;
  const int nbeg  = split * NCHUNK;

  // branch-free per-lane B-column recipe (sentinels: 32 -> 1.0, 33 -> 0.0)
  int l, l2;
  if (c == 0)          { l = 32; l2 = 32; }                 // 1*1   -> Z
  else if (c <= KK)    { l = c - 1; l2 = 32; }              // v_l*1 -> u
  else if (c < C_TOT)  { int cc = c - (KK + 1); l = cc >> 5; l2 = cc & 31; }
  else                 { l = 33; l2 = 33; }                 // 0*0   -> pad

  // one float4 per thread per panel: rn = row in chunk, q = 4-col group
  const int rn = tid >> 3;
  const int q4 = (tid & 7) * 4;

  v8f acc0 = {}, acc1 = {};
  for (int nb = nbeg; nb < nbeg + NCHUNK; nb += 32) {
    // ---- stage masked w^T panel (32 n x 32 m), b128 load + b128 LDS store
    {
      const float4 w4 = *(const float4*)&W[(size_t)(nb + rn) * MM + m0 + q4];
      float4 msk;
      msk.x = (w4.x >= W_EPS) ? w4.x : 0.f;
      msk.y = (w4.y >= W_EPS) ? w4.y : 0.f;
      msk.z = (w4.z >= W_EPS) ? w4.z : 0.f;
      msk.w = (w4.w >= W_EPS) ? w4.w : 0.f;
      *(float4*)&As[rn * LDS_STR + q4] = msk;
    }
    // ---- stage v panel (32 n x 32 l) + sentinel columns
    *(float4*)&Vs[rn * LDS_STR + q4] = *(const float4*)&v[(size_t)(nb + rn) * KK + q4];
    if (tid < 32) {
      Vs[tid * LDS_STR + 32] = 1.f;
      Vs[tid * LDS_STR + 33] = 0.f;
    }
    // prefetch next chunk while we compute on this one
    if (nb + 32 < nbeg + NCHUNK) {
      __builtin_prefetch(&W[(size_t)(nb + 32 + rn) * MM + m0 + q4], 0, 1);
      __builtin_prefetch(&v[(size_t)(nb + 32 + rn) * KK + q4], 0, 1);
    }
    __syncthreads();

    #pragma unroll
    for (int sub = 0; sub < 8; ++sub) {       // 8 x (16x16x4) steps per chunk
      const int r0 = sub * 4 + 2 * hi;        // this lane's K rows: r0, r0+1
      const float* vr0 = &Vs[(r0    ) * LDS_STR];
      const float* vr1 = &Vs[(r0 + 1) * LDS_STR];
      v2f b;  b.x  = vr0[l] * vr0[l2];  b.y  = vr1[l] * vr1[l2];
      v2f a0; a0.x = As[(r0) * LDS_STR + cl];      a0.y = As[(r0 + 1) * LDS_STR + cl];
      v2f a1; a1.x = As[(r0) * LDS_STR + 16 + cl]; a1.y = As[(r0 + 1) * LDS_STR + 16 + cl];
      acc0 = __builtin_amdgcn_wmma_f32_16x16x4_f32(false, a0, false, b,
                                                   (short)0, acc0, false, false);
      acc1 = __builtin_amdgcn_wmma_f32_16x16x4_f32(false, a1, false, b,
                                                   (short)0, acc1, false, false);
    }
    __syncthreads();
  }

  // 16x16 f32 C/D layout: VGPR r -> row r (lanes 0-15) / row r+8 (lanes 16-31)
  float* So = Sp + (size_t)split * MM * SC;
  #pragma unroll
  for (int r = 0; r < 8; ++r) {
    So[(size_t)(m0      + r + 8 * hi) * SC + c] = acc0[r];
    So[(size_t)(m0 + 16 + r + 8 * hi) * SC + c] = acc1[r];
  }
}

// ------------------------------------------------------------------
// Kernel 2b: deterministic fixed-order reduction of the split-K partials.
// ------------------------------------------------------------------
__global__ __launch_bounds__(256)
void reduce_kernel(const float* __restrict__ Sp, float* __restrict__ S) {
  const size_t i = (size_t)blockIdx.x * 256 + threadIdx.x;
  const size_t MSZ = (size_t)MM * SC;
  S[i] = ((Sp[i] + Sp[i + MSZ]) + (Sp[i + 2 * MSZ] + Sp[i + 3 * MSZ]));
}

// ------------------------------------------------------------------
// Kernel 3: count[m] = #{n : W[n,m] >= eps}. Coalesced over m; exact-integer
// fp atomics -> deterministic.
// ------------------------------------------------------------------
__global__ __launch_bounds__(256)
void count_kernel(const float* __restrict__ W, float* __restrict__ cnt) {
  const int mcol = threadIdx.x;
  const int n0 = blockIdx.x * 256;
  float cacc = 0.f;
  for (int n = n0; n < n0 + 256; ++n)
    cacc += (W[(size_t)n * MM + mcol] >= W_EPS) ? 1.f : 0.f;
  atomicAdd(&cnt[mcol], cacc);
}

// ------------------------------------------------------------------
// Kernel 4: psi[m] = tr(Op C Op^T)/Z - ||Op u||^2 / Z^2, gated on count>=2.
// One wave per m; lane k owns row k of omega_parent_m.
// ------------------------------------------------------------------
__global__ __launch_bounds__(32)
void finalize_kernel(const float* __restrict__ op, const float* __restrict__ S,
                     const float* __restrict__ cnt, float* __restrict__ out) {
  __shared__ float Os[32 * 33];
  __shared__ float Cs[32 * 32];
  __shared__ float us[32];
  const int m = blockIdx.x, tid = threadIdx.x;
  const float* Op = op + (size_t)m * (KK * KK);
  const float* Sm = S + (size_t)m * SC;

  #pragma unroll
  for (int i = 0; i < 32; ++i) Os[i * 33 + tid] = Op[i * 32 + tid];
  #pragma unroll
  for (int i = 0; i < 32; ++i) Cs[i * 32 + tid] = Sm[33 + i * 32 + tid];
  us[tid] = Sm[1 + tid];
  const float Z = Sm[0];
  __syncthreads();

  float t[32];
  #pragma unroll
  for (int l = 0; l < 32; ++l) t[l] = Os[tid * 33 + l];

  float r = 0.f;                                 // (Op u)_k
  #pragma unroll
  for (int l = 0; l < 32; ++l) r += t[l] * us[l];

  float q = 0.f;                                 // row_k Op * C * row_k Op^T
  #pragma unroll
  for (int j = 0; j < 32; ++j) {
    float a2 = 0.f;
    #pragma unroll
    for (int l = 0; l < 32; ++l) a2 += t[l] * Cs[l * 32 + j];  // broadcast reads
    q += a2 * t[j];
  }

  float s1 = r * r;
  for (int off = 16; off > 0; off >>= 1) {       // wave32 reductions
    s1 += __shfl_xor(s1, off, 32);
    q  += __shfl_xor(q,  off, 32);
  }
  if (tid == 0) {
    float Zs = fmaxf(Z, 1e-30f);
    float psi = q / Zs - s1 / (Zs * Zs);
    out[m] = (cnt[m] >= 1.5f) ? psi : 0.f;
  }
}

// ------------------------------------------------------------------
extern "C" void kernel_launch(void* const* d_in, const int* in_sizes, int n_in,
                              void* d_out, int out_size, void* d_ws, size_t ws_size,
                              hipStream_t stream) {
  const float* W  = (const float*)d_in[0];   // (N, M)
  const float* mu = (const float*)d_in[1];   // (N, K)
  const float* oc = (const float*)d_in[2];   // (N, K, K)
  const float* op = (const float*)d_in[3];   // (M, K, K)
  float* out = (float*)d_out;                // (M,)

  char* ws = (char*)d_ws;
  size_t off = 0;
  float* v   = (float*)(ws + off); off += (size_t)NN * KK * 4;            // 512 KB
  float* Sp  = (float*)(ws + off); off += (size_t)NSPLIT * MM * SC * 4;   // 4.7 MB
  float* S   = (float*)(ws + off); off += (size_t)MM * SC * 4;            // 1.2 MB
  float* cnt = (float*)(ws + off); off += MM * 4;                         // 1 KB

  hipMemsetAsync(cnt, 0, MM * sizeof(float), stream);
  solve_kernel      <<<NN, 32, 0, stream>>>(oc, mu, v);
  moment_gemm_kernel<<<dim3(MM / 32, SC / 128, NSPLIT), 256, 0, stream>>>(W, v, Sp);
  reduce_kernel     <<<(MM * SC) / 256, 256, 0, stream>>>(Sp, S);
  count_kernel      <<<NN / 256, 256, 0, stream>>>(W, cnt);
  finalize_kernel   <<<MM, 32, 0, stream>>>(op, S, cnt, out);
}